// MedianFilterWeighted3d_82265803588189
// MI455X (gfx1250) — compile-verified
//
#include <hip/hip_runtime.h>

// ---------------------------------------------------------------------------
// Weighted 3D median filter on [1,21,16,256,256] f32:
//   1) 5x5x5 replicate-padded box filter (separable):
//      - Kernel 1: W+H passes, LDS-tiled, halo loaded with CDNA5
//        global_load_async_to_lds_b32 (ASYNCcnt path), mp -> d_out
//      - Kernel 2: D pass as a 16x16 matmul (D==16) via V_WMMA_F32_16X16X4_F32
//   2) channel-normalize + cumsum + weighted-median selection fused into
//      kernel 2, in-place on d_out (disjoint tiles per block).
// ---------------------------------------------------------------------------

#define C_CH 21
#define DDIM 16
#define HDIM 256
#define WDIM 256
#define PLANE (HDIM * WDIM)   // 65536 elements per (c,d) slice

// WH tile geometry (kernel 1)
#define TW 64                 // output tile width
#define TH 16                 // output tile height
#define LW (TW + 4)           // 68: input tile width with halo
#define LH (TH + 4)           // 20: input tile height with halo

typedef __attribute__((ext_vector_type(2))) float v2f;
typedef __attribute__((ext_vector_type(8))) float v8f;

__device__ __forceinline__ int clampi(int v, int lo, int hi) {
    return v < lo ? lo : (v > hi ? hi : v);
}

// D-pass conv matrix: M[m][k] = (1/5) * #{ j in [-2,2] : clamp(m+j,0,15)==k }
__device__ __forceinline__ float dconv_wgt(int m, int k) {
    int cnt = 0;
#pragma unroll
    for (int j = -2; j <= 2; ++j) {
        int p = clampi(m + j, 0, DDIM - 1);
        cnt += (p == k) ? 1 : 0;
    }
    return 0.2f * (float)cnt;
}

// ---------------------------------------------------------------------------
// Kernel 1: separable W+H 5-tap box conv (scale 1/25), replicate padding.
// Grid: 336 planes * 4 w-tiles * 16 h-tiles = 21504 blocks, 256 threads.
// Halo tile is DMA'd into LDS with global_load_async_to_lds_b32.
// ---------------------------------------------------------------------------
__global__ __launch_bounds__(256) void wh_conv(const float* __restrict__ src,
                                               float* __restrict__ dst) {
    __shared__ float in_tile[LH * LW];   // 20*68*4 = 5440 B
    __shared__ float vs_tile[TH * LW];   // 16*68*4 = 4352 B

    const int tid   = threadIdx.x;
    const int bid   = blockIdx.x;
    const int plane = bid >> 6;          // c*16 + d, 0..335
    const int ht    = (bid & 63) >> 2;
    const int wt    = bid & 3;
    const int h0    = ht * TH;
    const int w0    = wt * TW;

    const float* sp = src + (size_t)plane * PLANE;
    const unsigned lds_base = (unsigned)(size_t)(&in_tile[0]);  // LDS byte offset

    // ---- Phase A: async halo-tile load (global -> LDS, no VGPR round trip)
    for (int li = tid; li < LH * LW; li += 256) {
        const int r  = li / LW;                    // 0..19
        const int c  = li - r * LW;                // 0..67
        const int gh = clampi(h0 + r - 2, 0, HDIM - 1);
        const int gw = clampi(w0 + c - 2, 0, WDIM - 1);
        const float* gp = sp + gh * WDIM + gw;
        const unsigned lo = lds_base + (unsigned)(li * 4);
        asm volatile("global_load_async_to_lds_b32 %0, %1, off"
                     :: "v"(lo), "v"(gp) : "memory");
    }
    asm volatile("s_wait_asynccnt 0x0" ::: "memory");
    __syncthreads();

    // ---- Phase B: vertical 5-sum into vs_tile
    for (int li = tid; li < TH * LW; li += 256) {
        const int r = li / LW;                     // 0..15
        const int c = li - r * LW;                 // 0..67
        const float s = in_tile[(r + 0) * LW + c] + in_tile[(r + 1) * LW + c] +
                        in_tile[(r + 2) * LW + c] + in_tile[(r + 3) * LW + c] +
                        in_tile[(r + 4) * LW + c];
        vs_tile[li] = s;
    }
    __syncthreads();

    // ---- Phase C: horizontal sliding 5-sum, 4 outputs/thread, float4 store
    {
        const int row = tid >> 4;                  // 0..15
        const int c0  = (tid & 15) << 2;           // 0..60
        const float* vr = &vs_tile[row * LW + c0];
        float v0 = vr[0], v1 = vr[1], v2 = vr[2], v3 = vr[3];
        float v4 = vr[4], v5 = vr[5], v6 = vr[6], v7 = vr[7];
        const float inv25 = 1.0f / 25.0f;
        float s = v0 + v1 + v2 + v3 + v4;
        float4 o;
        o.x = s * inv25;  s += v5 - v0;
        o.y = s * inv25;  s += v6 - v1;
        o.z = s * inv25;  s += v7 - v2;
        o.w = s * inv25;
        float* dp = dst + (size_t)plane * PLANE + (size_t)(h0 + row) * WDIM + w0 + c0;
        *(float4*)dp = o;                          // 16B aligned
    }
}

// ---------------------------------------------------------------------------
// Kernel 2: D-dim conv via WMMA f32 16x16x4 (K chained 4x) + channel median.
// Block: 256 threads (8 waves). Grid: HDIM * (WDIM/16) blocks.
// Wave `wv` computes channels wv, wv+8, wv+16 (<21): y-tile [16d x 16w] -> LDS.
// After barrier, thread t handles point (d=t/16, n=t%16): 21-channel median.
// In-place on `buf` (reads finish before the barrier; tiles are disjoint).
// ---------------------------------------------------------------------------
__global__ __launch_bounds__(256) void dconv_median(float* __restrict__ buf) {
    __shared__ float y_lds[C_CH][DDIM * 16];   // 21 * 256 * 4 = 21.5 KB

    const int tid  = threadIdx.x;
    const int wv   = tid >> 5;
    const int lane = tid & 31;
    const int h    = blockIdx.x >> 4;
    const int w0   = (blockIdx.x & 15) << 4;

    const int n  = lane & 15;      // column (W offset) for B/C fragments; row m for A
    const int hi = lane >> 4;      // lane half selects K pair (A/B) / M half (C)

    // ---- Phase 1: per-channel 16x16 D-conv tiles via V_WMMA_F32_16X16X4_F32
    for (int c = wv; c < C_CH; c += 8) {       // wave-uniform loop: EXEC stays full
        const float* src = buf + (size_t)c * DDIM * PLANE + (size_t)h * WDIM + w0;
        v8f acc = {};
#pragma unroll
        for (int kk = 0; kk < 4; ++kk) {
            const int k0 = kk * 4 + hi * 2;    // this lane's K pair within chunk
            v2f a, b;
            a.x = dconv_wgt(n, k0);
            a.y = dconv_wgt(n, k0 + 1);
            b.x = src[(size_t)k0 * PLANE + n];
            b.y = src[(size_t)(k0 + 1) * PLANE + n];
            acc = __builtin_amdgcn_wmma_f32_16x16x4_f32(
                false, a, false, b, (short)0, acc, false, false);
        }
        // C/D layout: VGPR v holds rows M=v (lanes 0-15) and M=v+8 (lanes 16-31)
#pragma unroll
        for (int v = 0; v < 8; ++v) {
            int d = v + hi * 8;
            y_lds[c][d * 16 + n] = acc[v];
        }
    }
    __syncthreads();

    // ---- Phase 2: normalize + cumsum + weighted-median channel selection
    {
        const int d  = tid >> 4;
        const int nn = tid & 15;

        float y[C_CH];
        float sum = 0.0f;
#pragma unroll
        for (int c = 0; c < C_CH; ++c) {
            y[c] = y_lds[c][tid];
            sum += y[c];
        }
        const float inv = 1.0f / sum;

        float s = 0.0f;
        int med0 = -1;         // last c with y>0 && cumsum<0.5, default 0
        int med1 = 0;          // first c>0 with cumsum>0.5, default 0
        bool set1 = false;
#pragma unroll
        for (int c = 0; c < C_CH; ++c) {
            float yn = y[c] * inv;
            s += yn;
            if (yn > 0.0f && s < 0.5f) med0 = c;
            if (!set1 && s > 0.5f && c > 0) { med1 = c; set1 = true; }
        }
        if (med0 < 0) med0 = 0;

#pragma unroll
        for (int c = 0; c < C_CH; ++c) {
            float o = (c == med0 || c == med1) ? y[c] * inv : 0.0f;
            buf[(((size_t)c * DDIM + d) * HDIM + h) * WDIM + w0 + nn] = o;
        }
    }
}

// ---------------------------------------------------------------------------
extern "C" void kernel_launch(void* const* d_in, const int* in_sizes, int n_in,
                              void* d_out, int out_size, void* d_ws, size_t ws_size,
                              hipStream_t stream) {
    const float* mp = (const float*)d_in[0];
    float* out = (float*)d_out;

    // 336 planes * (256/64) w-tiles * (256/16) h-tiles = 21504 blocks
    wh_conv<<<(C_CH * DDIM) * (WDIM / TW) * (HDIM / TH), 256, 0, stream>>>(mp, out);
    dconv_median<<<HDIM * (WDIM / 16), 256, 0, stream>>>(out);
    (void)d_ws; (void)ws_size; (void)n_in; (void)out_size; (void)in_sizes;
}